// GCN_41609643164181
// MI455X (gfx1250) — compile-verified
//
#include <hip/hip_runtime.h>
#include <hip/hip_bf16.h>

#define D_IN 128
#define D_HID 256
#define NC 7

typedef float v2f __attribute__((ext_vector_type(2)));
typedef float v8f __attribute__((ext_vector_type(8)));

// ---------------------------------------------------------------------------
// Kernel 1: zero the layer-1 accumulator, initialize d_out with b2 broadcast.
// ---------------------------------------------------------------------------
__global__ void gcn_init(float* __restrict__ agg, int n_agg,
                         float* __restrict__ out, int n_out,
                         const float* __restrict__ b2) {
    int i = blockIdx.x * blockDim.x + threadIdx.x;
    if (i < n_agg) agg[i] = 0.0f;
    if (i < n_out) out[i] = b2[i % NC];
}

// ---------------------------------------------------------------------------
// Kernel 2: layer-1 scatter.  agg[dst, :] += features[src, :]  (128 dims)
// One thread per (edge, 4-dim chunk): float4 gather + 4 f32 global atomics.
// Both source (25.6MB) and accumulator (25.6MB) are L2-resident (192MB L2).
// ---------------------------------------------------------------------------
__global__ void gcn_scatter1(const float* __restrict__ feat,
                             const int* __restrict__ esrc,
                             const int* __restrict__ edst,
                             float* __restrict__ agg, int n_edges) {
    long long tid = (long long)blockIdx.x * blockDim.x + threadIdx.x;
    int e  = (int)(tid >> 5);
    int d4 = ((int)tid & 31) << 2;
    if (e >= n_edges) return;
    int s = esrc[e];
    int d = edst[e];
    const float4 v = *reinterpret_cast<const float4*>(feat + (size_t)s * D_IN + d4);
    float* p = agg + (size_t)d * D_IN + d4;
    unsafeAtomicAdd(p + 0, v.x);
    unsafeAtomicAdd(p + 1, v.y);
    unsafeAtomicAdd(p + 2, v.z);
    unsafeAtomicAdd(p + 3, v.w);
}

// ---------------------------------------------------------------------------
// Kernel 3: fused  Y = relu(agg @ W1 + b1) @ W2   (no layer-2 bias here).
// One wave (32 threads) per 16-row tile.  The 16x16 H tile is produced by
// V_WMMA_F32_16X16X4_F32 (exact f32), pushed through LDS for the tiny
// [16,16]x[16,7] contraction with W2, accumulated in registers.
// H (50000x256, 51MB) never touches memory; only Y (50000x7) is written.
// ---------------------------------------------------------------------------
__global__ __launch_bounds__(32)
void gcn_gemm_fused(const float* __restrict__ agg,
                    const float* __restrict__ W1,
                    const float* __restrict__ b1,
                    const float* __restrict__ W2,
                    float* __restrict__ Y, int n_nodes) {
    __shared__ float lds[16 * 17];           // 16x16 tile, padded stride 17

    const int lane = threadIdx.x;            // 0..31 (wave32)
    const int half = lane >> 4;              // 0: K+{0,1} | 1: K+{2,3}
    const int l16  = lane & 15;              // A: row M ; B/C/D: col N
    const int m0   = blockIdx.x * 16;

    int mrow = m0 + l16;
    if (mrow >= n_nodes) mrow = n_nodes - 1; // clamp (dup row, store guarded)
    const float* arow = agg + (size_t)mrow * D_IN;

    float yacc[4] = {0.0f, 0.0f, 0.0f, 0.0f};

    for (int nb = 0; nb < D_HID; nb += 16) {
        v8f acc = {};
        // --- WMMA K-loop: H[m0:m0+16, nb:nb+16] = A[16,128] * W1[128,16] ---
        for (int k0 = 0; k0 < D_IN; k0 += 4) {
            v2f a, b;
            // A fragment: lane half selects K pair {k0,k0+1} / {k0+2,k0+3}
            a.x = arow[k0 + 2 * half + 0];
            a.y = arow[k0 + 2 * half + 1];
            // B fragment: row K striped across lanes (N = l16)
            const float* wcol = W1 + (size_t)(k0 + 2 * half) * D_HID + nb + l16;
            b.x = wcol[0];
            b.y = wcol[D_HID];
            acc = __builtin_amdgcn_wmma_f32_16x16x4_f32(
                false, a, false, b, (short)0, acc, false, false);
        }
        // --- bias + relu, spill 16x16 tile to LDS ---
        const float bn = b1[nb + l16];
#pragma unroll
        for (int r = 0; r < 8; ++r) {
            float h = acc[r] + bn;           // D layout: VGPR r -> M = r+8*half
            h = h > 0.0f ? h : 0.0f;
            lds[(r + 8 * half) * 17 + l16] = h;
        }
        __syncthreads();
        // --- tiny contraction with W2[nb:nb+16, 0:7]: 112 (M,c) pairs ---
#pragma unroll
        for (int j = 0; j < 4; ++j) {
            int idx = lane + 32 * j;
            if (idx < 16 * NC) {
                int M = idx / NC;
                int c = idx - M * NC;
                float s = 0.0f;
#pragma unroll
                for (int N = 0; N < 16; ++N)
                    s += lds[M * 17 + N] * W2[(size_t)(nb + N) * NC + c];
                yacc[j] += s;
            }
        }
        __syncthreads();
    }

    // --- write Y tile (exclusive rows, plain stores) ---
#pragma unroll
    for (int j = 0; j < 4; ++j) {
        int idx = lane + 32 * j;
        if (idx < 16 * NC) {
            int M = idx / NC;
            int c = idx - M * NC;
            int row = m0 + M;
            if (row < n_nodes) Y[(size_t)row * NC + c] = yacc[j];
        }
    }
}

// ---------------------------------------------------------------------------
// Kernel 4: layer-2 scatter in the *projected* 7-dim space:
//   out[dst, c] += Y[src, c]     (out pre-initialized to b2)
// 36x less atomic traffic than aggregating the 256-wide H.
// ---------------------------------------------------------------------------
__global__ void gcn_scatter2(const float* __restrict__ Y,
                             const int* __restrict__ esrc,
                             const int* __restrict__ edst,
                             float* __restrict__ out, int n_edges) {
    long long tid = (long long)blockIdx.x * blockDim.x + threadIdx.x;
    int e = (int)(tid >> 3);
    int c = (int)tid & 7;
    if (e >= n_edges || c >= NC) return;
    int s = esrc[e];
    int d = edst[e];
    unsafeAtomicAdd(out + (size_t)d * NC + c, Y[(size_t)s * NC + c]);
}

// ---------------------------------------------------------------------------
extern "C" void kernel_launch(void* const* d_in, const int* in_sizes, int n_in,
                              void* d_out, int out_size, void* d_ws, size_t ws_size,
                              hipStream_t stream) {
    const float* feat = (const float*)d_in[0];
    const int*   esrc = (const int*)d_in[1];
    const int*   edst = (const int*)d_in[2];
    const float* W1   = (const float*)d_in[3];
    const float* b1   = (const float*)d_in[4];
    const float* W2   = (const float*)d_in[5];
    const float* b2   = (const float*)d_in[6];
    float*       out  = (float*)d_out;

    const int n_nodes = in_sizes[0] / D_IN;
    const int n_edges = in_sizes[1];

    float* agg = (float*)d_ws;                        // [n_nodes, 128]
    float* Y   = agg + (size_t)n_nodes * D_IN;        // [n_nodes, 7]

    const int n_agg = n_nodes * D_IN;
    const int n_out = n_nodes * NC;

    // 1) init accumulator + output(bias)
    {
        int total = n_agg > n_out ? n_agg : n_out;
        gcn_init<<<(total + 255) / 256, 256, 0, stream>>>(agg, n_agg, out, n_out, b2);
    }
    // 2) layer-1 edge scatter (128-wide)
    {
        long long total = (long long)n_edges * 32;
        gcn_scatter1<<<(unsigned)((total + 255) / 256), 256, 0, stream>>>(
            feat, esrc, edst, agg, n_edges);
    }
    // 3) fused WMMA GEMM: Y = relu(agg@W1+b1) @ W2
    gcn_gemm_fused<<<(n_nodes + 15) / 16, 32, 0, stream>>>(agg, W1, b1, W2, Y, n_nodes);
    // 4) layer-2 edge scatter (7-wide, post-projection)
    {
        long long total = (long long)n_edges * 8;
        gcn_scatter2<<<(unsigned)((total + 255) / 256), 256, 0, stream>>>(
            Y, esrc, edst, out, n_edges);
    }
}